// GCN_9586367004794
// MI455X (gfx1250) — compile-verified
//
#include <hip/hip_runtime.h>
#include <hip/hip_bf16.h>
#include <math.h>

// ---------------------------------------------------------------------------
// Two-layer GCN for MI455X (gfx1250, wave32).
//   layer1: h1 = x @ W1;  agg1 = D^-1/2 A D^-1/2 h1 + b1;  r = relu(agg1)
//   layer2: (aggregate-first, linearity) t2 = D^-1/2 A D^-1/2 r;
//           out = t2 @ W2 + b2;  log_softmax rows
// GEMMs use V_WMMA_F32_16X16X4_F32 (exact fp32 matrix pipe).
// ---------------------------------------------------------------------------

typedef __attribute__((ext_vector_type(2))) float v2f;
typedef __attribute__((ext_vector_type(8))) float v8f;

#define FIN  512
#define FHID 16
#define FOUT 40
#define FPAD 48   // FOUT padded to 3 WMMA column tiles

static inline int cdiv(int a, int b) { return (a + b - 1) / b; }

// ---------------- degree / normalization --------------------------------
__global__ void deg_init_kernel(float* __restrict__ deg, int n) {
    int i = blockIdx.x * blockDim.x + threadIdx.x;
    if (i < n) deg[i] = 1.0f;                       // self loop
}

__global__ void deg_count_kernel(const int* __restrict__ col,
                                 float* __restrict__ deg, int e) {
    int i = blockIdx.x * blockDim.x + threadIdx.x;
    if (i < e) atomicAdd(&deg[col[i]], 1.0f);
}

__global__ void deg_rsqrt_kernel(float* __restrict__ deg, int n) {
    int i = blockIdx.x * blockDim.x + threadIdx.x;
    if (i < n) deg[i] = rsqrtf(deg[i]);             // deg -> dis in place
}

__global__ void norm_kernel(const int* __restrict__ row,
                            const int* __restrict__ col,
                            const float* __restrict__ dis,
                            float* __restrict__ nrm, int e) {
    int i = blockIdx.x * blockDim.x + threadIdx.x;
    if (i < e) nrm[i] = dis[row[i]] * dis[col[i]];
}

// ---------------- GEMM1: h1[N,16] = x[N,512] @ W1[512,16] ----------------
// 8 waves/block; each wave owns a 16-row M tile; W1 staged in LDS (32 KB).
__global__ __launch_bounds__(256) void gemm1_kernel(
        const float* __restrict__ x, const float* __restrict__ W1,
        float* __restrict__ h1, int n) {
    __shared__ float sW[FIN * FHID];
    for (int i = threadIdx.x; i < FIN * FHID; i += 256) sW[i] = W1[i];
    __syncthreads();

    const int wave    = threadIdx.x >> 5;
    const int lane    = threadIdx.x & 31;
    const int halfSel = lane >> 4;          // 0: lanes 0-15, 1: lanes 16-31
    const int l16     = lane & 15;
    const int mbase   = (blockIdx.x * 8 + wave) * 16;

    int arow = mbase + l16;                 // A row this lane streams
    if (arow >= n) arow = n - 1;            // clamp (stores predicated below)
    const float* aptr = x + (size_t)arow * FIN + halfSel * 2;

    v8f acc = {};
    for (int k = 0; k < FIN; k += 4) {
        // A 16x4 frag: lane<16 -> K=k,k+1 ; lane>=16 -> K=k+2,k+3 (row M=l16)
        v2f a = *(const v2f*)(aptr + k);
        // B 4x16 frag: VGPR0 = row K=k(+2), VGPR1 = row K=k+1(+3), col N=l16
        const int kb = k + halfSel * 2;
        v2f b;
        b.x = sW[kb * FHID + l16];
        b.y = sW[(kb + 1) * FHID + l16];
        acc = __builtin_amdgcn_wmma_f32_16x16x4_f32(
                false, a, false, b, (short)0, acc, false, false);
    }
    // C/D layout: VGPR r -> (M = r + 8*halfSel, N = l16)
#pragma unroll
    for (int r = 0; r < 8; ++r) {
        int m = mbase + r + 8 * halfSel;
        if (m < n) h1[(size_t)m * FHID + l16] = acc[r];
    }
}

// ---------------- aggregation, layer 1 ----------------------------------
__global__ void agg1_init_kernel(const float* __restrict__ h1,
                                 const float* __restrict__ dis,
                                 const float* __restrict__ b1,
                                 float* __restrict__ agg, int n) {
    int idx = blockIdx.x * blockDim.x + threadIdx.x;
    if (idx < n * FHID) {
        int i = idx >> 4, c = idx & 15;
        float d = dis[i];
        agg[idx] = b1[c] + d * d * h1[idx];   // bias + self-loop term
    }
}

__global__ void agg1_edge_kernel(const int* __restrict__ row,
                                 const int* __restrict__ col,
                                 const float* __restrict__ nrm,
                                 const float* __restrict__ h1,
                                 float* __restrict__ agg, int e) {
    int idx = blockIdx.x * blockDim.x + threadIdx.x;
    if (idx < e * FHID) {
        int ei = idx >> 4, c = idx & 15;
        int r = row[ei], t = col[ei];
        atomicAdd(&agg[(size_t)t * FHID + c], nrm[ei] * h1[(size_t)r * FHID + c]);
    }
}

// ---------------- aggregation of relu(agg1), layer 2 (pre-transform) -----
__global__ void agg2_init_kernel(const float* __restrict__ agg1,
                                 const float* __restrict__ dis,
                                 float* __restrict__ t2, int n) {
    int idx = blockIdx.x * blockDim.x + threadIdx.x;
    if (idx < n * FHID) {
        int i = idx >> 4;
        float d = dis[i];
        t2[idx] = d * d * fmaxf(agg1[idx], 0.0f);   // relu fused
    }
}

__global__ void agg2_edge_kernel(const int* __restrict__ row,
                                 const int* __restrict__ col,
                                 const float* __restrict__ nrm,
                                 const float* __restrict__ agg1,
                                 float* __restrict__ t2, int e) {
    int idx = blockIdx.x * blockDim.x + threadIdx.x;
    if (idx < e * FHID) {
        int ei = idx >> 4, c = idx & 15;
        int r = row[ei], t = col[ei];
        atomicAdd(&t2[(size_t)t * FHID + c],
                  nrm[ei] * fmaxf(agg1[(size_t)r * FHID + c], 0.0f));
    }
}

// ---------------- GEMM2: out[N,40] = t2[N,16] @ W2[16,40] + b2 -----------
// N padded to 48 (3 WMMA column tiles); W2/b2 zero-padded in LDS.
__global__ __launch_bounds__(256) void gemm2_kernel(
        const float* __restrict__ t2, const float* __restrict__ W2,
        const float* __restrict__ b2, float* __restrict__ out, int n) {
    __shared__ float sW[FHID * FPAD];
    __shared__ float sB[FPAD];
    for (int i = threadIdx.x; i < FHID * FPAD; i += 256) {
        int r = i / FPAD, c = i % FPAD;
        sW[i] = (c < FOUT) ? W2[r * FOUT + c] : 0.0f;
    }
    if (threadIdx.x < FPAD)
        sB[threadIdx.x] = (threadIdx.x < FOUT) ? b2[threadIdx.x] : 0.0f;
    __syncthreads();

    const int wave    = threadIdx.x >> 5;
    const int lane    = threadIdx.x & 31;
    const int halfSel = lane >> 4;
    const int l16     = lane & 15;
    const int mbase   = (blockIdx.x * 8 + wave) * 16;

    int arow = mbase + l16;
    if (arow >= n) arow = n - 1;
    const float* aptr = t2 + (size_t)arow * FHID + halfSel * 2;

    v2f afrag[4];                              // K = 0..15, reused by 3 tiles
#pragma unroll
    for (int kk = 0; kk < 4; ++kk) afrag[kk] = *(const v2f*)(aptr + 4 * kk);

#pragma unroll
    for (int nt = 0; nt < 3; ++nt) {
        v8f acc = {};
#pragma unroll
        for (int kk = 0; kk < 4; ++kk) {
            const int kb = 4 * kk + halfSel * 2;
            v2f b;
            b.x = sW[kb * FPAD + nt * 16 + l16];
            b.y = sW[(kb + 1) * FPAD + nt * 16 + l16];
            acc = __builtin_amdgcn_wmma_f32_16x16x4_f32(
                    false, afrag[kk], false, b, (short)0, acc, false, false);
        }
        const int colx = nt * 16 + l16;
        if (colx < FOUT) {
            const float bias = sB[colx];
#pragma unroll
            for (int r = 0; r < 8; ++r) {
                int m = mbase + r + 8 * halfSel;
                if (m < n) out[(size_t)m * FOUT + colx] = acc[r] + bias;
            }
        }
    }
}

// ---------------- log_softmax (wave per row, wave32 shuffles) ------------
__global__ __launch_bounds__(256) void logsoftmax_kernel(float* __restrict__ out,
                                                         int n) {
    const int wave = threadIdx.x >> 5;
    const int lane = threadIdx.x & 31;
    const int rowi = blockIdx.x * 8 + wave;
    if (rowi >= n) return;                       // wave-uniform exit
    float* p = out + (size_t)rowi * FOUT;
    float v0 = p[lane];                          // cols 0..31
    float v1 = (lane < 8) ? p[lane + 32] : -INFINITY;   // cols 32..39
    float m = fmaxf(v0, v1);
#pragma unroll
    for (int off = 16; off >= 1; off >>= 1)
        m = fmaxf(m, __shfl_xor(m, off, 32));
    float s = __expf(v0 - m) + ((lane < 8) ? __expf(v1 - m) : 0.0f);
#pragma unroll
    for (int off = 16; off >= 1; off >>= 1)
        s += __shfl_xor(s, off, 32);
    float lse = m + __logf(s);
    p[lane] = v0 - lse;
    if (lane < 8) p[lane + 32] = v1 - lse;
}

// ---------------------------------------------------------------------------
extern "C" void kernel_launch(void* const* d_in, const int* in_sizes, int n_in,
                              void* d_out, int out_size, void* d_ws, size_t ws_size,
                              hipStream_t stream) {
    const float* x   = (const float*)d_in[0];
    const int*   ei  = (const int*)d_in[1];      // integer -> int32 per harness
    const float* W1  = (const float*)d_in[2];
    const float* b1  = (const float*)d_in[3];
    const float* W2  = (const float*)d_in[4];
    const float* b2  = (const float*)d_in[5];
    float* out = (float*)d_out;

    const int n = in_sizes[0] / FIN;             // 100000
    const int e = in_sizes[1] / 2;               // 3200000
    const int* erow = ei;                        // source nodes
    const int* ecol = ei + e;                    // target nodes

    // workspace carve-up (floats, 1 KB aligned chunks); total ~32 MB
    float* ws   = (float*)d_ws;
    float* dis  = ws;                                        // n (deg->dis)
    float* nrm  = dis  + (size_t)((n + 255) & ~255);         // e
    float* h1   = nrm  + (size_t)((e + 255) & ~255);         // n*16
    float* agg1 = h1   + (size_t)((n * FHID + 255) & ~255);  // n*16
    float* t2   = agg1 + (size_t)((n * FHID + 255) & ~255);  // n*16

    const int T = 256;
    deg_init_kernel <<<cdiv(n, T), T, 0, stream>>>(dis, n);
    deg_count_kernel<<<cdiv(e, T), T, 0, stream>>>(ecol, dis, e);
    deg_rsqrt_kernel<<<cdiv(n, T), T, 0, stream>>>(dis, n);
    norm_kernel     <<<cdiv(e, T), T, 0, stream>>>(erow, ecol, dis, nrm, e);

    gemm1_kernel    <<<cdiv(n, 128), T, 0, stream>>>(x, W1, h1, n);

    agg1_init_kernel<<<cdiv(n * FHID, T), T, 0, stream>>>(h1, dis, b1, agg1, n);
    agg1_edge_kernel<<<cdiv(e * FHID, T), T, 0, stream>>>(erow, ecol, nrm, h1, agg1, e);

    agg2_init_kernel<<<cdiv(n * FHID, T), T, 0, stream>>>(agg1, dis, t2, n);
    agg2_edge_kernel<<<cdiv(e * FHID, T), T, 0, stream>>>(erow, ecol, nrm, agg1, t2, e);

    gemm2_kernel    <<<cdiv(n, 128), T, 0, stream>>>(t2, W2, b2, out, n);

    logsoftmax_kernel<<<cdiv(n, 8), T, 0, stream>>>(out, n);
}